// StarTransformerEncoder_52956946759746
// MI455X (gfx1250) — compile-verified
//
#include <hip/hip_runtime.h>
#include <hip/hip_bf16.h>
#include <math.h>

// ---------------------------------------------------------------------------
// Star-Transformer encoder forward for MI455X (gfx1250, wave32).
// fp32 end-to-end; all large GEMMs use V_WMMA_F32_16X16X4_F32 (one wave owns a
// 16x16 output tile, K chained in steps of 4 per the CDNA5 A/B VGPR layout).
// ---------------------------------------------------------------------------

typedef float v2f __attribute__((ext_vector_type(2)));
typedef float v8f __attribute__((ext_vector_type(8)));

#define B_   4
#define L_   2048
#define D_   128
#define D2_  512
#define NH_  4
#define HD_  128
#define NL_  3
#define E_   300
#define LP1_ 2049
#define SCALE_ 0.08838834764831845f   /* 1/sqrt(128) */

// ---------------------------------------------------------------------------
// Generic WMMA f32 GEMM:  out[m,n] = epilogue( sum_k A[m,k]*W[n,k] + bias[n] )
//   mode 0: out = gemm + bias
//   mode 1: out = resid + leaky_relu(gemm + bias)      (slope 0.01)
// One wave per 16x16 tile; blockDim.x = 128 (4 waves). M may be non-mult-of-16.
// N, K must be multiples of 16 / 4 respectively (true for all call sites).
// ---------------------------------------------------------------------------
__global__ void gemm_wmma_f32(const float* __restrict__ A,
                              const float* __restrict__ W,
                              const float* __restrict__ bias,
                              const float* __restrict__ resid,
                              float* __restrict__ out,
                              int M, int N, int K, int mode) {
  int warp = threadIdx.x >> 5;
  int lane = threadIdx.x & 31;
  int Ntiles = N >> 4;
  int Mtiles = (M + 15) >> 4;
  int t = blockIdx.x * 4 + warp;
  if (t >= Mtiles * Ntiles) return;           // wave-uniform exit (EXEC all-1 for WMMA)
  int mt = t / Ntiles, nt = t % Ntiles;
  int row = lane & 15, hi = lane >> 4;        // hi selects K pair {0,1} vs {2,3}

  int m = mt * 16 + row;
  float amask = (m < M) ? 1.0f : 0.0f;
  const float* arow = A + (size_t)((m < M) ? m : 0) * K;
  const float* wrow = W + (size_t)(nt * 16 + row) * K;

  v8f c = {0.f, 0.f, 0.f, 0.f, 0.f, 0.f, 0.f, 0.f};
  for (int k0 = 0; k0 < K; k0 += 4) {
    if ((k0 & 63) == 0) {                     // global_prefetch_b8 ahead
      __builtin_prefetch(arow + k0 + 64, 0, 1);
      __builtin_prefetch(wrow + k0 + 64, 0, 1);
    }
    int kk = k0 + hi * 2;
    v2f a, b;
    a.x = arow[kk] * amask;  a.y = arow[kk + 1] * amask;
    b.x = wrow[kk];          b.y = wrow[kk + 1];
    c = __builtin_amdgcn_wmma_f32_16x16x4_f32(false, a, false, b,
                                              (short)0, c, false, false);
  }

  int n = nt * 16 + (lane & 15);
  float bn = bias[n];
#pragma unroll
  for (int r = 0; r < 8; ++r) {
    int mm = mt * 16 + r + hi * 8;
    if (mm < M) {
      size_t idx = (size_t)mm * N + n;
      float vv = c[r] + bn;
      if (mode == 1) {
        vv = (vv > 0.0f) ? vv : 0.01f * vv;
        out[idx] = resid[idx] + vv;
      } else {
        out[idx] = vv;
      }
    }
  }
}

// ---------------------------------------------------------------------------
// Embedding-gather GEMM: x[m,n] = sum_k emb[tokens[m]][k] * proj_w[n][k] + b[n]
// M = B*L = 8192, N = 128, K = 300. Writes x into two buffers (nodes, r_embs).
// ---------------------------------------------------------------------------
__global__ void embed_gemm_wmma(const int* __restrict__ tokens,
                                const float* __restrict__ emb,
                                const float* __restrict__ W,
                                const float* __restrict__ bias,
                                float* __restrict__ x0,
                                float* __restrict__ x1) {
  const int Ntiles = D_ >> 4;                 // 8
  const int tiles = (B_ * L_ / 16) * Ntiles;  // 4096
  int warp = threadIdx.x >> 5;
  int lane = threadIdx.x & 31;
  int t = blockIdx.x * 4 + warp;
  if (t >= tiles) return;
  int mt = t / Ntiles, nt = t % Ntiles;
  int row = lane & 15, hi = lane >> 4;

  int m = mt * 16 + row;
  const float* arow = emb + (size_t)tokens[m] * E_;
  const float* wrow = W + (size_t)(nt * 16 + row) * E_;

  v8f c = {0.f, 0.f, 0.f, 0.f, 0.f, 0.f, 0.f, 0.f};
  for (int k0 = 0; k0 < E_; k0 += 4) {        // 300 = 75 * 4
    int kk = k0 + hi * 2;
    v2f a, b;
    a.x = arow[kk];  a.y = arow[kk + 1];
    b.x = wrow[kk];  b.y = wrow[kk + 1];
    c = __builtin_amdgcn_wmma_f32_16x16x4_f32(false, a, false, b,
                                              (short)0, c, false, false);
  }

  int n = nt * 16 + (lane & 15);
  float bn = bias[n];
#pragma unroll
  for (int r = 0; r < 8; ++r) {
    int mm = mt * 16 + r + hi * 8;
    size_t idx = (size_t)mm * D_ + n;
    float vv = c[r] + bn;
    x0[idx] = vv;
    x1[idx] = vv;
  }
}

// relay[b][d] = mean over l of x[b][l][d]
__global__ void relay_mean(const float* __restrict__ x, float* __restrict__ relay) {
  int idx = blockIdx.x * blockDim.x + threadIdx.x;
  if (idx >= B_ * D_) return;
  int b = idx >> 7, d = idx & 127;
  const float* p = x + (size_t)b * L_ * D_ + d;
  float s = 0.0f;
  for (int l = 0; l < L_; ++l) s += p[(size_t)l * D_];
  relay[idx] = s * (1.0f / (float)L_);
}

// LayerNorm over D=128; one wave32 per row, 4 elems/lane, shfl_xor reductions
__global__ void layernorm128(const float* __restrict__ xin,
                             const float* __restrict__ g,
                             const float* __restrict__ bta,
                             float* __restrict__ xout) {
  int row = blockIdx.x;
  int lane = threadIdx.x;
  const float* p = xin + (size_t)row * D_;
  float vals[4], s = 0.0f, sq = 0.0f;
#pragma unroll
  for (int j = 0; j < 4; ++j) {
    float t = p[lane + 32 * j];
    vals[j] = t; s += t; sq += t * t;
  }
#pragma unroll
  for (int o = 16; o >= 1; o >>= 1) {
    s  += __shfl_xor(s,  o, 32);
    sq += __shfl_xor(sq, o, 32);
  }
  float mu = s * (1.0f / 128.0f);
  float var = sq * (1.0f / 128.0f) - mu * mu;
  float rs = rsqrtf(var + 1e-5f);
#pragma unroll
  for (int j = 0; j < 4; ++j) {
    int d = lane + 32 * j;
    xout[(size_t)row * D_ + d] = (vals[j] - mu) * rs * g[d] + bta[d];
  }
}

// Tiny projection for B=4 rows: out[b][n] = act(sum_k in[b][k]*W[n][k] + b[n])
__global__ void vec_proj(const float* __restrict__ in,
                         const float* __restrict__ W,
                         const float* __restrict__ bias,
                         float* __restrict__ out, int N, int K, int act) {
  int idx = blockIdx.x * blockDim.x + threadIdx.x;
  if (idx >= B_ * N) return;
  int b = idx / N, n = idx % N;
  const float* irow = in + (size_t)b * K;
  const float* wrow = W + (size_t)n * K;
  float s = bias[n];
  for (int k = 0; k < K; ++k) s += irow[k] * wrow[k];
  if (act) s = (s > 0.0f) ? s : 0.01f * s;
  out[(size_t)b * N + n] = s;
}

// Ring (windowed) attention: 5 keys = {l-1,l,l+1 (zero-padded), ak_emb, ak_relay}
// One wave32 per (b,l); loops heads; 4 contiguous floats per lane per head.
__global__ void ring_attn(const float* __restrict__ q,
                          const float* __restrict__ k,
                          const float* __restrict__ v,
                          const float* __restrict__ ak,
                          const float* __restrict__ av,
                          const float* __restrict__ akr,
                          const float* __restrict__ avr,
                          float* __restrict__ att) {
  int bl = blockIdx.x;
  int b = bl >> 11, l = bl & (L_ - 1);
  int lane = threadIdx.x;
  for (int h = 0; h < NH_; ++h) {
    int hoff = h * HD_ + lane * 4;
    float4 qv = *(const float4*)(q + (size_t)bl * D2_ + hoff);
    float e[5];
    float mx = -__builtin_inff();
#pragma unroll
    for (int u = 0; u < 5; ++u) {
      float4 kk = {0.f, 0.f, 0.f, 0.f};
      if (u < 3) {
        int pos = l + u - 1;
        if (pos >= 0 && pos < L_)
          kk = *(const float4*)(k + (size_t)(b * L_ + pos) * D2_ + hoff);
      } else if (u == 3) {
        kk = *(const float4*)(ak + (size_t)bl * D2_ + hoff);
      } else {
        kk = *(const float4*)(akr + (size_t)b * D2_ + hoff);
      }
      float p = qv.x * kk.x + qv.y * kk.y + qv.z * kk.z + qv.w * kk.w;
#pragma unroll
      for (int o = 16; o >= 1; o >>= 1) p += __shfl_xor(p, o, 32);
      e[u] = p * SCALE_;
      mx = fmaxf(mx, e[u]);
    }
    float den = 0.0f;
#pragma unroll
    for (int u = 0; u < 5; ++u) { e[u] = __expf(e[u] - mx); den += e[u]; }
    float inv = 1.0f / den;
    float4 acc = {0.f, 0.f, 0.f, 0.f};
#pragma unroll
    for (int u = 0; u < 5; ++u) {
      float4 vv = {0.f, 0.f, 0.f, 0.f};
      if (u < 3) {
        int pos = l + u - 1;
        if (pos >= 0 && pos < L_)
          vv = *(const float4*)(v + (size_t)(b * L_ + pos) * D2_ + hoff);
      } else if (u == 3) {
        vv = *(const float4*)(av + (size_t)bl * D2_ + hoff);
      } else {
        vv = *(const float4*)(avr + (size_t)b * D2_ + hoff);
      }
      float w = e[u] * inv;
      acc.x += w * vv.x; acc.y += w * vv.y; acc.z += w * vv.z; acc.w += w * vv.w;
    }
    *(float4*)(att + (size_t)bl * D2_ + hoff) = acc;
  }
}

// y[b][0] = relay[b]; y[b][1+l] = nodes[b][l]
__global__ void build_y(const float* __restrict__ relay,
                        const float* __restrict__ nodes,
                        float* __restrict__ y) {
  int idx = blockIdx.x * blockDim.x + threadIdx.x;
  if (idx >= B_ * LP1_ * D_) return;
  int d = idx & 127;
  int r = (idx >> 7) % LP1_;
  int b = idx / (LP1_ * D_);
  y[idx] = (r == 0) ? relay[b * D_ + d]
                    : nodes[((size_t)b * L_ + (r - 1)) * D_ + d];
}

// Relay (global) attention over L+1 keys per (b, head); masked softmax.
// Block 256 threads per (b,h); scores staged in LDS.
__global__ void relay_attn(const float* __restrict__ sq,
                           const float* __restrict__ skb,
                           const float* __restrict__ svb,
                           const int* __restrict__ mask,
                           float* __restrict__ satt) {
  __shared__ float sc[LP1_];
  __shared__ float red[256];
  int bh = blockIdx.x;
  int b = bh >> 2, h = bh & 3;
  int tid = threadIdx.x;
  const float* qrow = sq + (size_t)b * D2_ + h * HD_;

  for (int j = tid; j < LP1_; j += 256) {
    const float* krow = skb + (size_t)(b * LP1_ + j) * D2_ + h * HD_;
    float s = 0.0f;
    for (int d = 0; d < HD_; ++d) s += qrow[d] * krow[d];
    s *= SCALE_;
    if (j > 0 && mask[b * L_ + (j - 1)] == 0) s = -__builtin_inff();
    sc[j] = s;
  }
  __syncthreads();

  float m = -__builtin_inff();
  for (int j = tid; j < LP1_; j += 256) m = fmaxf(m, sc[j]);
  red[tid] = m; __syncthreads();
  for (int s2 = 128; s2 > 0; s2 >>= 1) {
    if (tid < s2) red[tid] = fmaxf(red[tid], red[tid + s2]);
    __syncthreads();
  }
  float mx = red[0];
  __syncthreads();

  float ps = 0.0f;
  for (int j = tid; j < LP1_; j += 256) {
    float e = __expf(sc[j] - mx);
    sc[j] = e; ps += e;
  }
  __syncthreads();
  red[tid] = ps; __syncthreads();
  for (int s2 = 128; s2 > 0; s2 >>= 1) {
    if (tid < s2) red[tid] += red[tid + s2];
    __syncthreads();
  }
  float inv = 1.0f / red[0];
  __syncthreads();

  if (tid < HD_) {
    float acc = 0.0f;
    for (int j = 0; j < LP1_; ++j)
      acc += sc[j] * svb[(size_t)(b * LP1_ + j) * D2_ + h * HD_ + tid];
    satt[(size_t)b * D2_ + h * HD_ + tid] = acc * inv;
  }
}

// nodes = where(pad, 0, nodes)
__global__ void mask_nodes(float* __restrict__ nodes, const int* __restrict__ mask) {
  int idx = blockIdx.x * blockDim.x + threadIdx.x;
  if (idx >= B_ * L_ * D_) return;
  if (mask[idx >> 7] == 0) nodes[idx] = 0.0f;
}

// ---------------------------------------------------------------------------
extern "C" void kernel_launch(void* const* d_in, const int* in_sizes, int n_in,
                              void* d_out, int out_size, void* d_ws, size_t ws_size,
                              hipStream_t stream) {
  const int*   tokens = (const int*)  d_in[0];
  const int*   maskp  = (const int*)  d_in[1];
  const float* emb    = (const float*)d_in[2];
  const float* proj_w = (const float*)d_in[3];
  const float* proj_b = (const float*)d_in[4];
  const float* ng     = (const float*)d_in[5];
  const float* nb     = (const float*)d_in[6];
  const float* rq_w = (const float*)d_in[7],  *rq_b = (const float*)d_in[8];
  const float* rk_w = (const float*)d_in[9],  *rk_b = (const float*)d_in[10];
  const float* rv_w = (const float*)d_in[11], *rv_b = (const float*)d_in[12];
  const float* ro_w = (const float*)d_in[13], *ro_b = (const float*)d_in[14];
  const float* sq_w = (const float*)d_in[15], *sq_b = (const float*)d_in[16];
  const float* sk_w = (const float*)d_in[17], *sk_b = (const float*)d_in[18];
  const float* sv_w = (const float*)d_in[19], *sv_b = (const float*)d_in[20];
  const float* so_w = (const float*)d_in[21], *so_b = (const float*)d_in[22];

  float* ws = (float*)d_ws;
  const size_t BL  = (size_t)B_ * L_;
  const size_t BLP = (size_t)B_ * LP1_;
  size_t o = 0;
  float* r_embs = ws + o; o += BL * D_;
  float* nodes  = ws + o; o += BL * D_;
  float* nx     = ws + o; o += BL * D_;
  float* qb     = ws + o; o += BL * D2_;
  float* kb     = ws + o; o += BL * D2_;
  float* vb     = ws + o; o += BL * D2_;
  float* akb    = ws + o; o += BL * D2_;
  float* avb    = ws + o; o += BL * D2_;
  float* attb   = ws + o; o += BL * D2_;
  float* yb     = ws + o; o += BLP * D_;
  float* skb    = ws + o; o += BLP * D2_;
  float* svb    = ws + o; o += BLP * D2_;
  float* akr    = ws + o; o += B_ * D2_;
  float* avr    = ws + o; o += B_ * D2_;
  float* sqv    = ws + o; o += B_ * D2_;
  float* satt   = ws + o; o += B_ * D2_;
  float* relay  = ws + o; o += B_ * D_;

  // ---- embedding + projection, relay = mean ----
  {
    int tiles = (B_ * L_ / 16) * (D_ / 16);           // 4096
    embed_gemm_wmma<<<(tiles + 3) / 4, 128, 0, stream>>>(
        tokens, emb, proj_w, proj_b, nodes, r_embs);
    relay_mean<<<(B_ * D_ + 255) / 256, 256, 0, stream>>>(nodes, relay);
  }

  const float* nullres = nullptr;
  for (int i = 0; i < NL_; ++i) {
    const float* rqW = rq_w + (size_t)i * D2_ * D_;  const float* rqB = rq_b + (size_t)i * D2_;
    const float* rkW = rk_w + (size_t)i * D2_ * D_;  const float* rkB = rk_b + (size_t)i * D2_;
    const float* rvW = rv_w + (size_t)i * D2_ * D_;  const float* rvB = rv_b + (size_t)i * D2_;
    const float* roW = ro_w + (size_t)i * D_ * D2_;  const float* roB = ro_b + (size_t)i * D_;
    const float* sqW = sq_w + (size_t)i * D2_ * D_;  const float* sqB = sq_b + (size_t)i * D2_;
    const float* skW = sk_w + (size_t)i * D2_ * D_;  const float* skB = sk_b + (size_t)i * D2_;
    const float* svW = sv_w + (size_t)i * D2_ * D_;  const float* svB = sv_b + (size_t)i * D2_;
    const float* soW = so_w + (size_t)i * D_ * D2_;  const float* soB = so_b + (size_t)i * D_;

    // nx = LN(nodes)
    layernorm128<<<B_ * L_, 32, 0, stream>>>(nodes, ng + i * D_, nb + i * D_, nx);

    // q, k, v from nx; ak, av from r_embs (M=8192, N=512, K=128)
    {
      int tiles = (B_ * L_ / 16) * (D2_ / 16);        // 16384
      int grid = (tiles + 3) / 4;
      gemm_wmma_f32<<<grid, 128, 0, stream>>>(nx,     rqW, rqB, nullres, qb,  B_*L_, D2_, D_, 0);
      gemm_wmma_f32<<<grid, 128, 0, stream>>>(nx,     rkW, rkB, nullres, kb,  B_*L_, D2_, D_, 0);
      gemm_wmma_f32<<<grid, 128, 0, stream>>>(nx,     rvW, rvB, nullres, vb,  B_*L_, D2_, D_, 0);
      gemm_wmma_f32<<<grid, 128, 0, stream>>>(r_embs, rkW, rkB, nullres, akb, B_*L_, D2_, D_, 0);
      gemm_wmma_f32<<<grid, 128, 0, stream>>>(r_embs, rvW, rvB, nullres, avb, B_*L_, D2_, D_, 0);
    }
    // relay-derived ring keys/values (B=4 rows)
    vec_proj<<<(B_ * D2_ + 255) / 256, 256, 0, stream>>>(relay, rkW, rkB, akr, D2_, D_, 0);
    vec_proj<<<(B_ * D2_ + 255) / 256, 256, 0, stream>>>(relay, rvW, rvB, avr, D2_, D_, 0);

    // windowed ring attention
    ring_attn<<<B_ * L_, 32, 0, stream>>>(qb, kb, vb, akb, avb, akr, avr, attb);

    // nodes += leaky_relu(att @ roW.T + roB)   (M=8192, N=128, K=512)
    {
      int tiles = (B_ * L_ / 16) * (D_ / 16);         // 4096
      gemm_wmma_f32<<<(tiles + 3) / 4, 128, 0, stream>>>(
          attb, roW, roB, nodes, nodes, B_*L_, D_, D2_, 1);
    }

    // y = [relay; nodes] ; sk, sv over L+1 (M=8196, N=512, K=128)
    build_y<<<((int)(BLP * D_) + 255) / 256, 256, 0, stream>>>(relay, nodes, yb);
    {
      int Mtiles = (B_ * LP1_ + 15) / 16;             // 513
      int tiles = Mtiles * (D2_ / 16);                // 16416
      int grid = (tiles + 3) / 4;
      gemm_wmma_f32<<<grid, 128, 0, stream>>>(yb, skW, skB, nullres, skb, B_*LP1_, D2_, D_, 0);
      gemm_wmma_f32<<<grid, 128, 0, stream>>>(yb, svW, svB, nullres, svb, B_*LP1_, D2_, D_, 0);
    }
    vec_proj<<<(B_ * D2_ + 255) / 256, 256, 0, stream>>>(relay, sqW, sqB, sqv, D2_, D_, 0);

    // global attention -> satt; relay = leaky_relu(satt @ soW.T + soB)
    relay_attn<<<B_ * NH_, 256, 0, stream>>>(sqv, skb, svb, maskp, satt);
    vec_proj<<<(B_ * D_ + 255) / 256, 256, 0, stream>>>(satt, soW, soB, relay, D_, D2_, 1);

    // nodes = where(pad, 0, nodes)
    mask_nodes<<<((int)(BL * D_) + 255) / 256, 256, 0, stream>>>(nodes, maskp);
  }

  hipMemcpyAsync(d_out, relay, (size_t)B_ * D_ * sizeof(float),
                 hipMemcpyDeviceToDevice, stream);
}